// GatedSpectralClassification_51513837748774
// MI455X (gfx1250) — compile-verified
//
#include <hip/hip_runtime.h>
#include <hip/hip_bf16.h>

// ---------------------------------------------------------------------------
// GatedSpectralClassification on gfx1250 (MI455X), f32 end-to-end.
// All large matmuls run on V_WMMA_F32_16X16X4_F32 (native f32 matrix pipe).
// Double-buffered LDS staging with b128 global loads.
// ---------------------------------------------------------------------------

typedef float v2f __attribute__((ext_vector_type(2)));
typedef float v4f __attribute__((ext_vector_type(4)));
typedef float v8f __attribute__((ext_vector_type(8)));

#define B_  16
#define S_  1024
#define H_  512
#define C_  7
#define BS_ (B_ * S_)
#define EPS_ 1e-10f
#define LPAD 20   // row stride in dwords (16 + 4): 16B-aligned rows, no bank conflicts

static __device__ __forceinline__ v8f wmma4(v2f a, v2f b, v8f c) {
    // 8-arg form: (neg_a, A, neg_b, B, c_mod, C, reuse_a, reuse_b)
    return __builtin_amdgcn_wmma_f32_16x16x4_f32(
        false, a, false, b, (short)0, c, false, false);
}

// ---------------------------------------------------------------------------
// Small helper: block-wide sum (blockDim.x == 256, wave32)
// ---------------------------------------------------------------------------
static __device__ float blockReduceSum256(float v) {
    __shared__ float red[8];
    #pragma unroll
    for (int off = 16; off > 0; off >>= 1) v += __shfl_xor(v, off, 32);
    const int warp = threadIdx.x >> 5, lane = threadIdx.x & 31;
    __syncthreads();                 // protect red[] from previous call
    if (lane == 0) red[warp] = v;
    __syncthreads();
    float r = 0.f;
    #pragma unroll
    for (int w = 0; w < 8; ++w) r += red[w];
    return r;
}

// ---------------------------------------------------------------------------
// Weff[i][j] = gW[i][j] + gW[H+i][j] - 0.5*gW[2H+i][j]   (i,j in [0,512))
// ---------------------------------------------------------------------------
__global__ __launch_bounds__(256)
void weff_kernel(const float* __restrict__ gW, float* __restrict__ weff) {
    int idx = blockIdx.x * blockDim.x + threadIdx.x;
    if (idx >= H_ * H_) return;
    weff[idx] = gW[idx] + gW[H_ * H_ + idx] - 0.5f * gW[2 * H_ * H_ + idx];
}

// ---------------------------------------------------------------------------
// One block per (b,s) row: sum_all = t+a+v, and per-modality entropy over H.
// ---------------------------------------------------------------------------
__global__ __launch_bounds__(256)
void ent_sum_kernel(const float* __restrict__ t, const float* __restrict__ a,
                    const float* __restrict__ v, float* __restrict__ sum_all,
                    float* __restrict__ entT, float* __restrict__ entA,
                    float* __restrict__ entV) {
    const size_t row = blockIdx.x;
    const size_t base = row * H_;
    const int i0 = threadIdx.x, i1 = threadIdx.x + 256;

    float t0 = t[base + i0], t1 = t[base + i1];
    float a0 = a[base + i0], a1 = a[base + i1];
    float v0 = v[base + i0], v1 = v[base + i1];

    sum_all[base + i0] = t0 + a0 + v0;
    sum_all[base + i1] = t1 + a1 + v1;

    float st = blockReduceSum256(fabsf(t0) + fabsf(t1));
    float sa = blockReduceSum256(fabsf(a0) + fabsf(a1));
    float sv = blockReduceSum256(fabsf(v0) + fabsf(v1));

    float rt = 1.f / (st + EPS_), ra = 1.f / (sa + EPS_), rv = 1.f / (sv + EPS_);
    float f;
    float et = 0.f, ea = 0.f, ev = 0.f;
    f = fabsf(t0) * rt; et += f * __logf(f + EPS_);
    f = fabsf(t1) * rt; et += f * __logf(f + EPS_);
    f = fabsf(a0) * ra; ea += f * __logf(f + EPS_);
    f = fabsf(a1) * ra; ea += f * __logf(f + EPS_);
    f = fabsf(v0) * rv; ev += f * __logf(f + EPS_);
    f = fabsf(v1) * rv; ev += f * __logf(f + EPS_);

    et = blockReduceSum256(et);
    ea = blockReduceSum256(ea);
    ev = blockReduceSum256(ev);
    if (threadIdx.x == 0) {
        entT[row] = -et; entA[row] = -ea; entV[row] = -ev;
    }
}

// ---------------------------------------------------------------------------
// Row degree + rsqrt for fused adjacency; one wave per row (8 rows / block).
// ---------------------------------------------------------------------------
__global__ __launch_bounds__(256)
void rowsum_dsi_kernel(const float* __restrict__ sp, const float* __restrict__ tm,
                       float* __restrict__ dsi) {
    const int row  = blockIdx.x * 8 + (threadIdx.x >> 5);   // 0..BS-1
    const int lane = threadIdx.x & 31;
    const int iloc = row & (S_ - 1);
    const size_t base = (size_t)row * S_;
    float s = 0.f;
    #pragma unroll 4
    for (int j = lane; j < S_; j += 32) {
        float aij = 0.5f * (sp[base + j] + tm[base + j]);
        if (j == iloc) aij += 1.f;
        s += aij;
    }
    #pragma unroll
    for (int off = 16; off > 0; off >>= 1) s += __shfl_xor(s, off, 32);
    if (lane == 0) dsi[row] = rsqrtf(s + EPS_);
}

// ---------------------------------------------------------------------------
// norm_adj[b,i,j] = dsi[b,i] * (0.5*(sp+tm)+(i==j)) * dsi[b,j]
// ---------------------------------------------------------------------------
__global__ __launch_bounds__(256)
void normadj_kernel(const float* __restrict__ sp, const float* __restrict__ tm,
                    const float* __restrict__ dsi, float* __restrict__ na) {
    size_t idx = (size_t)blockIdx.x * blockDim.x + threadIdx.x;
    const size_t total = (size_t)B_ * S_ * S_;
    if (idx >= total) return;
    size_t b = idx >> 20;             // / (S*S)
    size_t r = idx & ((1u << 20) - 1);
    size_t i = r >> 10, j = r & (S_ - 1);
    float aij = 0.5f * (sp[idx] + tm[idx]) + ((i == j) ? 1.f : 0.f);
    na[idx] = dsi[b * S_ + i] * aij * dsi[b * S_ + j];
}

// ---------------------------------------------------------------------------
// Tiled f32 WMMA GEMM: C = op(A[M,K] * B[K,N]).  Tile 128x64, BK=16.
// 8 waves -> 4x2 wave grid, each wave owns 2x2 16x16 accumulators.
// Double-buffered LDS; staging via b128 global loads into registers.
// EPI 0: store  | 1: *0.5 store | 2: sigmoid-gate epilogue | 3: bias+ReLU
// Batched via blockIdx.z with element strides.
// ---------------------------------------------------------------------------
template <int EPI>
__global__ __launch_bounds__(256)
void gemm_wmma_f32(const float* __restrict__ A, const float* __restrict__ Bm,
                   float* __restrict__ C, int M, int N, int K,
                   int lda, int ldb, int ldc,
                   long strideA, long strideB, long strideC,
                   const float* __restrict__ e0,   // P (EPI2) / bias (EPI3)
                   const float* __restrict__ e1,   // ent row vector (EPI2)
                   const float* __restrict__ e2,   // gw_last col vector (EPI2)
                   const float* __restrict__ e3,   // gate_b (EPI2)
                   const float* __restrict__ e4) { // feat (EPI2)
    __shared__ float As[2][128][LPAD];
    __shared__ float Bs[2][64][LPAD];

    const float* Ab = A + (size_t)blockIdx.z * strideA;
    const float* Bb = Bm + (size_t)blockIdx.z * strideB;
    float*       Cb = C + (size_t)blockIdx.z * strideC;
    const int grow0 = (int)blockIdx.z * M;          // global row base (epilogue)

    const int bm = blockIdx.y * 128;
    const int bn = blockIdx.x * 64;
    const int tid = threadIdx.x;
    const int warp = tid >> 5, lane = tid & 31;
    const int waveM = warp & 3, waveN = warp >> 2;  // 4x2 wave grid
    const int half = lane >> 4, mlan = lane & 15;

    // ---- staging geometry (per thread: 2 float4 of A, 1 float4 of B) ----
    const int amRow = tid >> 2;           // 0..63 (and +64)
    const int akQ   = (tid & 3) * 4;      // k quad within 16
    const int bkRow = tid >> 4;           // 0..15
    const int bnQ   = (tid & 15) * 4;     // n quad within 64
    const float* aP0 = Ab + (size_t)(bm + amRow) * lda + akQ;
    const float* aP1 = Ab + (size_t)(bm + amRow + 64) * lda + akQ;
    const float* bP  = Bb + (size_t)bkRow * ldb + bn + bnQ;
    const size_t bStep = (size_t)ldb * 16;

    v8f acc[2][2];
    #pragma unroll
    for (int i = 0; i < 2; ++i)
        #pragma unroll
        for (int j = 0; j < 2; ++j)
            #pragma unroll
            for (int e = 0; e < 8; ++e) acc[i][j][e] = 0.f;

    v4f ra0, ra1, rb;
    const int ktiles = K >> 4;

    // prologue: fetch + stage tile 0
    ra0 = *(const v4f*)(aP0);
    ra1 = *(const v4f*)(aP1);
    rb  = *(const v4f*)(bP);
    *(v4f*)&As[0][amRow][akQ]      = ra0;
    *(v4f*)&As[0][amRow + 64][akQ] = ra1;
    Bs[0][bnQ + 0][bkRow] = rb[0];
    Bs[0][bnQ + 1][bkRow] = rb[1];
    Bs[0][bnQ + 2][bkRow] = rb[2];
    Bs[0][bnQ + 3][bkRow] = rb[3];
    __syncthreads();

    for (int kt = 0; kt < ktiles; ++kt) {
        const int cur = kt & 1, nxt = cur ^ 1;
        // issue next tile's global loads before touching LDS (latency hiding)
        if (kt + 1 < ktiles) {
            ra0 = *(const v4f*)(aP0 + (size_t)(kt + 1) * 16);
            ra1 = *(const v4f*)(aP1 + (size_t)(kt + 1) * 16);
            rb  = *(const v4f*)(bP + (size_t)(kt + 1) * bStep);
        }
        // compute on current buffer
        #pragma unroll
        for (int kk4 = 0; kk4 < 4; ++kk4) {
            const int kfrag = kk4 * 4 + half * 2;   // even -> 8B aligned
            v2f a0 = *(const v2f*)&As[cur][waveM * 32 +  0 + mlan][kfrag];
            v2f a1 = *(const v2f*)&As[cur][waveM * 32 + 16 + mlan][kfrag];
            v2f b0 = *(const v2f*)&Bs[cur][waveN * 32 +  0 + mlan][kfrag];
            v2f b1 = *(const v2f*)&Bs[cur][waveN * 32 + 16 + mlan][kfrag];
            acc[0][0] = wmma4(a0, b0, acc[0][0]);
            acc[0][1] = wmma4(a0, b1, acc[0][1]);
            acc[1][0] = wmma4(a1, b0, acc[1][0]);
            acc[1][1] = wmma4(a1, b1, acc[1][1]);
        }
        // stage next tile into the other buffer (safe: it was last read in
        // iteration kt-1, and the barrier at the end of kt-1 fenced that)
        if (kt + 1 < ktiles) {
            *(v4f*)&As[nxt][amRow][akQ]      = ra0;
            *(v4f*)&As[nxt][amRow + 64][akQ] = ra1;
            Bs[nxt][bnQ + 0][bkRow] = rb[0];
            Bs[nxt][bnQ + 1][bkRow] = rb[1];
            Bs[nxt][bnQ + 2][bkRow] = rb[2];
            Bs[nxt][bnQ + 3][bkRow] = rb[3];
        }
        __syncthreads();
    }

    // ---- epilogue ----
    #pragma unroll
    for (int mi = 0; mi < 2; ++mi) {
        #pragma unroll
        for (int ni = 0; ni < 2; ++ni) {
            const int tM = bm + waveM * 32 + mi * 16;
            const int tN = bn + waveN * 32 + ni * 16;
            #pragma unroll
            for (int i = 0; i < 8; ++i) {
                const int row = tM + (half ? i + 8 : i);
                const int col = tN + mlan;
                float val = acc[mi][ni][i];
                if (EPI == 1) {
                    val *= 0.5f;
                } else if (EPI == 2) {
                    const size_t grow = (size_t)(grow0 + row);
                    float logit = val + e0[grow * H_ + col]
                                + e1[grow] * e2[col] + e3[col];
                    float g = 1.f / (1.f + __expf(-logit));
                    val = g * e4[grow * H_ + col];
                } else if (EPI == 3) {
                    val = fmaxf(val + e0[col], 0.f);
                }
                Cb[(size_t)row * ldc + col] = val;
            }
        }
    }
}

// ---------------------------------------------------------------------------
// Classifier: out[row,c] = h[row,:] . Wc[:,c] + bc[c]   (N=7, VALU is fine)
// ---------------------------------------------------------------------------
__global__ __launch_bounds__(256)
void classifier_kernel(const float* __restrict__ h, const float* __restrict__ Wc,
                       const float* __restrict__ bc, float* __restrict__ out) {
    int idx = blockIdx.x * blockDim.x + threadIdx.x;
    if (idx >= BS_ * C_) return;
    int row = idx / C_, c = idx - row * C_;
    const float* hr = h + (size_t)row * H_;
    float acc = bc[c];
    #pragma unroll 8
    for (int k = 0; k < H_; ++k) acc = fmaf(hr[k], Wc[k * C_ + c], acc);
    out[idx] = acc;
}

// ---------------------------------------------------------------------------
extern "C" void kernel_launch(void* const* d_in, const int* in_sizes, int n_in,
                              void* d_out, int out_size, void* d_ws, size_t ws_size,
                              hipStream_t stream) {
    const float* text   = (const float*)d_in[0];
    const float* audio  = (const float*)d_in[1];
    const float* visual = (const float*)d_in[2];
    const float* sp     = (const float*)d_in[3];
    const float* tm     = (const float*)d_in[4];
    // d_in[5] = qmask (unused by reference math)
    const float* gate_W = (const float*)d_in[6];
    const float* gate_b = (const float*)d_in[7];
    const float* W0 = (const float*)d_in[8];  const float* b0 = (const float*)d_in[9];
    const float* W1 = (const float*)d_in[10]; const float* b1 = (const float*)d_in[11];
    const float* W2 = (const float*)d_in[12]; const float* b2 = (const float*)d_in[13];
    const float* Wc = (const float*)d_in[14]; const float* bc = (const float*)d_in[15];
    float* out = (float*)d_out;

    // ---- workspace layout (floats) ----
    float* ws    = (float*)d_ws;
    float* weff  = ws;                         // 512*512
    float* entT  = weff + H_ * H_;             // 16384 each
    float* entA  = entT + BS_;
    float* entV  = entA + BS_;
    float* dsi   = entV + BS_;                 // 16384
    float* sumA  = dsi + BS_;                  // 16384*512  (later reused: tmp)
    float* P     = sumA + (size_t)BS_ * H_;    // 16384*512  (later reused: h)
    float* h0    = P    + (size_t)BS_ * H_;    // 16384*1536
    float* na    = h0   + (size_t)BS_ * 3 * H_;// 16*1024*1024 (L2-resident)
    float* tmp   = sumA;                       // free after P GEMM
    float* hbuf  = P;                          // free after gating GEMMs

    const float* gWc_rows = gate_W + (size_t)2 * H_ * H_;  // rows [2H,3H)
    const float* gw_last  = gate_W + (size_t)3 * H_ * H_;  // row 3H

    // 1. Weff = Wa + Wb - 0.5*Wc_g
    weff_kernel<<<(H_ * H_ + 255) / 256, 256, 0, stream>>>(gate_W, weff);

    // 2. sum_all + per-modality entropies
    ent_sum_kernel<<<BS_, 256, 0, stream>>>(text, audio, visual, sumA,
                                            entT, entA, entV);

    // 3. P = 0.5 * sum_all @ Wc_g            [16384,512]x[512,512]
    dim3 gP(H_ / 64, BS_ / 128, 1);
    gemm_wmma_f32<1><<<gP, 256, 0, stream>>>(sumA, gWc_rows, P,
        BS_, H_, H_, H_, H_, H_, 0, 0, 0, nullptr, nullptr, nullptr, nullptr, nullptr);

    // 4. Gated fusion -> h0 columns [m*H,(m+1)*H)
    gemm_wmma_f32<2><<<gP, 256, 0, stream>>>(text, weff, h0 + 0 * H_,
        BS_, H_, H_, H_, H_, 3 * H_, 0, 0, 0, P, entT, gw_last, gate_b, text);
    gemm_wmma_f32<2><<<gP, 256, 0, stream>>>(audio, weff, h0 + 1 * H_,
        BS_, H_, H_, H_, H_, 3 * H_, 0, 0, 0, P, entA, gw_last, gate_b, audio);
    gemm_wmma_f32<2><<<gP, 256, 0, stream>>>(visual, weff, h0 + 2 * H_,
        BS_, H_, H_, H_, H_, 3 * H_, 0, 0, 0, P, entV, gw_last, gate_b, visual);

    // 5-6. normalized adjacency (67 MB -> stays in 192 MB L2 for 3 reuses)
    rowsum_dsi_kernel<<<BS_ / 8, 256, 0, stream>>>(sp, tm, dsi);
    normadj_kernel<<<(B_ * S_ * S_ + 255) / 256, 256, 0, stream>>>(sp, tm, dsi, na);

    // 7. Spectral layer 0: tmp = h0 @ W0 ; hbuf = relu(na @ tmp + b0)
    gemm_wmma_f32<0><<<gP, 256, 0, stream>>>(h0, W0, tmp,
        BS_, H_, 3 * H_, 3 * H_, H_, H_, 0, 0, 0,
        nullptr, nullptr, nullptr, nullptr, nullptr);
    dim3 gS(H_ / 64, S_ / 128, B_);
    gemm_wmma_f32<3><<<gS, 256, 0, stream>>>(na, tmp, hbuf,
        S_, H_, S_, S_, H_, H_,
        (long)S_ * S_, (long)S_ * H_, (long)S_ * H_,
        b0, nullptr, nullptr, nullptr, nullptr);

    // 8. Layer 1
    gemm_wmma_f32<0><<<gP, 256, 0, stream>>>(hbuf, W1, tmp,
        BS_, H_, H_, H_, H_, H_, 0, 0, 0,
        nullptr, nullptr, nullptr, nullptr, nullptr);
    gemm_wmma_f32<3><<<gS, 256, 0, stream>>>(na, tmp, hbuf,
        S_, H_, S_, S_, H_, H_,
        (long)S_ * S_, (long)S_ * H_, (long)S_ * H_,
        b1, nullptr, nullptr, nullptr, nullptr);

    // 9. Layer 2
    gemm_wmma_f32<0><<<gP, 256, 0, stream>>>(hbuf, W2, tmp,
        BS_, H_, H_, H_, H_, H_, 0, 0, 0,
        nullptr, nullptr, nullptr, nullptr, nullptr);
    gemm_wmma_f32<3><<<gS, 256, 0, stream>>>(na, tmp, hbuf,
        S_, H_, S_, S_, H_, H_,
        (long)S_ * S_, (long)S_ * H_, (long)S_ * H_,
        b2, nullptr, nullptr, nullptr, nullptr);

    // 10. Classifier
    classifier_kernel<<<(BS_ * C_ + 255) / 256, 256, 0, stream>>>(hbuf, Wc, bc, out);
}